// GCN_2542620639281
// MI455X (gfx1250) — compile-verified
//
#include <hip/hip_runtime.h>
#include <math.h>

#define TPB 256

typedef float v2f __attribute__((ext_vector_type(2)));
typedef float v8f __attribute__((ext_vector_type(8)));

// ---------------------------------------------------------------------------
// Utility kernels
// ---------------------------------------------------------------------------
__global__ void fill_zero_kernel(float* __restrict__ dst, long long n) {
    long long i = (long long)blockIdx.x * TPB + threadIdx.x;
    if (i < n) dst[i] = 0.0f;
}

__global__ void neg_copy_kernel(float* __restrict__ dst, const float* __restrict__ src,
                                long long n) {
    long long i = (long long)blockIdx.x * TPB + threadIdx.x;
    if (i < n) dst[i] = -src[i];
}

// deg[row[e]] += 1
__global__ void deg_kernel(const int* __restrict__ rowi, float* __restrict__ deg, int E) {
    int e = blockIdx.x * TPB + threadIdx.x;
    if (e < E) atomicAdd(&deg[rowi[e]], 1.0f);
}

// in-place: deg -> dinv = (deg>0) ? rsqrt(max(deg,1e-12)) : 0
__global__ void dinv_kernel(float* __restrict__ deg, int Nn) {
    int i = blockIdx.x * TPB + threadIdx.x;
    if (i < Nn) {
        float d = deg[i];
        deg[i] = (d > 0.0f) ? rsqrtf(fmaxf(d, 1e-12f)) : 0.0f;
    }
}

// norm[e] = -dinv[row]*dinv[col]
__global__ void norm_kernel(const int* __restrict__ rowi, const int* __restrict__ coli,
                            const float* __restrict__ dinv, float* __restrict__ nrm, int E) {
    int e = blockIdx.x * TPB + threadIdx.x;
    if (e < E) nrm[e] = -dinv[rowi[e]] * dinv[coli[e]];
}

// dst[row] += scale * norm[e] * src[col] ; edge-parallel, float4 feature chunks
__global__ void spmv_edge_kernel(const int* __restrict__ rowi, const int* __restrict__ coli,
                                 const float* __restrict__ nrm, const float* __restrict__ src,
                                 float* __restrict__ dst, int F4, float scale, long long total) {
    long long idx = (long long)blockIdx.x * TPB + threadIdx.x;
    if (idx >= total) return;
    int e = (int)(idx / F4);
    int c = (int)(idx - (long long)e * F4);
    float w = scale * nrm[e];
    const float4* s = (const float4*)(src + (long long)coli[e] * (F4 * 4));
    float4 v = s[c];
    float* d = dst + (long long)rowi[e] * (F4 * 4) + c * 4;
    atomicAdd(d + 0, w * v.x);
    atomicAdd(d + 1, w * v.y);
    atomicAdd(d + 2, w * v.z);
    atomicAdd(d + 3, w * v.w);
}

// ---------------------------------------------------------------------------
// WMMA f32 16x16x4 GEMM: C = (mode&1 ? C : 0) + A @ B  [+bias, relu if mode&2]
// A: [N, Kdim] row-major, B: [Kdim, Fout] row-major, C: [N, Fout]
// grid = (N/16, Fout/16), block = 32 (one wave, EXEC all-1s, no divergence)
//
// Fragment layouts per CDNA5 ISA 7.12.2:
//   A 16x4 f32:  lane(lo)=M, vgpr v holds K = k0 + 2*hi + v         (hi = lane>>4)
//   B 4x16 f32:  lane(lo)=N, vgpr v holds K = k0 + 2*hi + v
//   C/D 16x16:   lane(lo)=N, vgpr v holds M = v + 8*hi
// ---------------------------------------------------------------------------
__global__ __launch_bounds__(32)
void cheb_wmma_gemm(const float* __restrict__ A, const float* __restrict__ B,
                    const float* __restrict__ bias, float* __restrict__ C,
                    int Kdim, int Fout, int mode) {
    const int lane = threadIdx.x;
    const int hi = lane >> 4;
    const int lo = lane & 15;
    const long long rbase = (long long)blockIdx.x * 16;
    const int cc = blockIdx.y * 16 + lo;

    v8f acc = {};
    if (mode & 1) {
#pragma unroll
        for (int v = 0; v < 8; ++v)
            acc[v] = C[(rbase + v + 8 * hi) * Fout + cc];
    }

    const float* Arow = A + (rbase + lo) * (long long)Kdim;
    for (int k0 = 0; k0 < Kdim; k0 += 4) {
        float2 av = *(const float2*)(Arow + k0 + 2 * hi);
        v2f a;
        a[0] = av.x;
        a[1] = av.y;
        v2f b;
        b[0] = B[(long long)(k0 + 2 * hi) * Fout + cc];
        b[1] = B[(long long)(k0 + 2 * hi + 1) * Fout + cc];
        acc = __builtin_amdgcn_wmma_f32_16x16x4_f32(false, a, false, b, (short)0, acc,
                                                    false, false);
    }

#pragma unroll
    for (int v = 0; v < 8; ++v) {
        float val = acc[v];
        if (mode & 2) {
            val += bias[cc];
            val = fmaxf(val, 0.0f);
        }
        C[(rbase + v + 8 * hi) * Fout + cc] = val;
    }
}

// ---------------------------------------------------------------------------
// Mean pool + linear head + log_softmax
// ---------------------------------------------------------------------------
__global__ void pool_accum_kernel(const float* __restrict__ h, const int* __restrict__ batch,
                                  float* __restrict__ pooled, float* __restrict__ cnt,
                                  int Nn, int F) {
    long long idx = (long long)blockIdx.x * TPB + threadIdx.x;
    if (idx >= (long long)Nn * F) return;
    int i = (int)(idx / F);
    int f = (int)(idx - (long long)i * F);
    int g = batch[i];
    atomicAdd(&pooled[(long long)g * F + f], h[(long long)i * F + f]);
    if (f == 0) atomicAdd(&cnt[g], 1.0f);
}

__global__ void head_kernel(const float* __restrict__ pooled, const float* __restrict__ cnt,
                            const float* __restrict__ lw, const float* __restrict__ lb,
                            float* __restrict__ out, int Gg, int F) {
    int g = blockIdx.x * TPB + threadIdx.x;
    if (g >= Gg) return;
    float c = fmaxf(cnt[g], 1.0f);
    float z0 = lb[0], z1 = lb[1];
    for (int f = 0; f < F; ++f) {
        float p = pooled[(long long)g * F + f] / c;
        z0 += p * lw[2 * f + 0];
        z1 += p * lw[2 * f + 1];
    }
    float m = fmaxf(z0, z1);
    float lse = m + logf(expf(z0 - m) + expf(z1 - m));
    out[2 * g + 0] = z0 - lse;
    out[2 * g + 1] = z1 - lse;
}

// ---------------------------------------------------------------------------
// Host orchestration
// ---------------------------------------------------------------------------
extern "C" void kernel_launch(void* const* d_in, const int* in_sizes, int n_in,
                              void* d_out, int out_size, void* d_ws, size_t ws_size,
                              hipStream_t stream) {
    const float* x = (const float*)d_in[0];
    const int* edge = (const int*)d_in[1];
    const int* batch = (const int*)d_in[2];
    const int E = in_sizes[1] / 2;
    const int Nn = in_sizes[2];
    const int Gg = out_size / 2;
    const int Fin0 = in_sizes[0] / Nn; // 116

    const int FinA[5] = {Fin0, 32, 32, 64, 64};
    const int FoutA[5] = {32, 32, 64, 64, 128};
    const float* Wp[5] = {(const float*)d_in[3], (const float*)d_in[5], (const float*)d_in[7],
                          (const float*)d_in[9], (const float*)d_in[11]};
    const float* bp[5] = {(const float*)d_in[4], (const float*)d_in[6], (const float*)d_in[8],
                          (const float*)d_in[10], (const float*)d_in[12]};
    const float* lin_w = (const float*)d_in[13];
    const float* lin_b = (const float*)d_in[14];
    float* out = (float*)d_out;

    const int* rowi = edge;
    const int* coli = edge + E;

    // --- carve workspace ---
    char* wptr = (char*)d_ws;
    auto alloc = [&](size_t bytes) -> float* {
        float* p = (float*)wptr;
        wptr += (bytes + 255) & ~(size_t)255;
        return p;
    };
    float* dinv = alloc((size_t)Nn * 4);
    float* nrm = alloc((size_t)E * 4);
    size_t tbytes = (size_t)Nn * Fin0 * 4; // Fin0 is the max feature width of any T
    float* tA = alloc(tbytes);
    float* tB = alloc(tbytes);
    float* tC = alloc(tbytes);
    float* h0 = alloc((size_t)Nn * 128 * 4);
    float* h1 = alloc((size_t)Nn * 128 * 4);
    float* pooled = alloc((size_t)Gg * 128 * 4);
    float* cnt = alloc((size_t)Gg * 4);
    (void)ws_size;
    (void)n_in;

    auto nb = [](long long n) { return (unsigned)((n + TPB - 1) / TPB); };

    // --- normalization: deg -> dinv (in place) -> per-edge norm ---
    fill_zero_kernel<<<nb(Nn), TPB, 0, stream>>>(dinv, Nn);
    deg_kernel<<<nb(E), TPB, 0, stream>>>(rowi, dinv, E);
    dinv_kernel<<<nb(Nn), TPB, 0, stream>>>(dinv, Nn);
    norm_kernel<<<nb(E), TPB, 0, stream>>>(rowi, coli, dinv, nrm, E);

    // --- 5 ChebConv layers ---
    const float* hin = x;
    float* houts[5] = {h0, h1, h0, h1, h0};
    for (int L = 0; L < 5; ++L) {
        const int Fi = FinA[L], Fo = FoutA[L];
        float* hout = houts[L];
        dim3 ggrid(Nn / 16, Fo / 16);
        const long long nT = (long long)Nn * Fi;
        const long long eTot = (long long)E * (Fi / 4);
        const size_t wstride = (size_t)Fi * Fo;

        // k=0: out = T0 @ W0
        cheb_wmma_gemm<<<ggrid, 32, 0, stream>>>(hin, Wp[L], bp[L], hout, Fi, Fo, 0);

        // T1 = Lx(T0)
        fill_zero_kernel<<<nb(nT), TPB, 0, stream>>>(tA, nT);
        spmv_edge_kernel<<<nb(eTot), TPB, 0, stream>>>(rowi, coli, nrm, hin, tA, Fi / 4, 1.0f, eTot);
        cheb_wmma_gemm<<<ggrid, 32, 0, stream>>>(tA, Wp[L] + wstride, bp[L], hout, Fi, Fo, 1);

        // T2 = 2*Lx(T1) - T0
        neg_copy_kernel<<<nb(nT), TPB, 0, stream>>>(tB, hin, nT);
        spmv_edge_kernel<<<nb(eTot), TPB, 0, stream>>>(rowi, coli, nrm, tA, tB, Fi / 4, 2.0f, eTot);
        cheb_wmma_gemm<<<ggrid, 32, 0, stream>>>(tB, Wp[L] + 2 * wstride, bp[L], hout, Fi, Fo, 1);

        // T3 = 2*Lx(T2) - T1
        neg_copy_kernel<<<nb(nT), TPB, 0, stream>>>(tC, tA, nT);
        spmv_edge_kernel<<<nb(eTot), TPB, 0, stream>>>(rowi, coli, nrm, tB, tC, Fi / 4, 2.0f, eTot);
        cheb_wmma_gemm<<<ggrid, 32, 0, stream>>>(tC, Wp[L] + 3 * wstride, bp[L], hout, Fi, Fo, 1);

        // T4 = 2*Lx(T3) - T2   (reuse tA; T1 dead by now)
        neg_copy_kernel<<<nb(nT), TPB, 0, stream>>>(tA, tB, nT);
        spmv_edge_kernel<<<nb(eTot), TPB, 0, stream>>>(rowi, coli, nrm, tC, tA, Fi / 4, 2.0f, eTot);
        // final k: accumulate + bias + ReLU
        cheb_wmma_gemm<<<ggrid, 32, 0, stream>>>(tA, Wp[L] + 4 * wstride, bp[L], hout, Fi, Fo, 3);

        hin = hout;
    }

    // --- global mean pool + head ---
    fill_zero_kernel<<<nb((long long)Gg * 128), TPB, 0, stream>>>(pooled, (long long)Gg * 128);
    fill_zero_kernel<<<nb(Gg), TPB, 0, stream>>>(cnt, Gg);
    pool_accum_kernel<<<nb((long long)Nn * 128), TPB, 0, stream>>>(hin, batch, pooled, cnt, Nn, 128);
    head_kernel<<<nb(Gg), TPB, 0, stream>>>(pooled, cnt, lin_w, lin_b, out, Gg, 128);
}